// GATEncoder_14207751815313
// MI455X (gfx1250) — compile-verified
//
#include <hip/hip_runtime.h>

typedef float v2f __attribute__((ext_vector_type(2)));
typedef float v8f __attribute__((ext_vector_type(8)));

#define N_NODES 50000
#define N_EDGES 800000

__device__ __forceinline__ float lrelu(float x) { return x > 0.f ? x : 0.2f * x; }

// ---------------------------------------------------------------------------
// CSR construction
// ---------------------------------------------------------------------------
__global__ __launch_bounds__(256) void k_zero2(int* a, int* b, int n) {
  int i = blockIdx.x * blockDim.x + threadIdx.x;
  if (i < n) { a[i] = 0; b[i] = 0; }
}

__global__ __launch_bounds__(256) void k_deg(const int* __restrict__ dst,
                                             int* __restrict__ deg, int e) {
  int i = blockIdx.x * blockDim.x + threadIdx.x;
  if (i < e) atomicAdd(&deg[dst[i]], 1);
}

__global__ __launch_bounds__(1024) void k_scan(const int* __restrict__ deg,
                                               int* __restrict__ rowptr, int n) {
  __shared__ int part[1024];
  int t = threadIdx.x;
  int per = (n + 1023) / 1024;
  int beg = t * per; if (beg > n) beg = n;
  int end = beg + per; if (end > n) end = n;
  int s = 0;
  for (int i = beg; i < end; ++i) s += deg[i];
  part[t] = s;
  __syncthreads();
  for (int off = 1; off < 1024; off <<= 1) {
    int v = 0;
    if (t >= off) v = part[t - off];
    __syncthreads();
    part[t] += v;
    __syncthreads();
  }
  int run = (t == 0) ? 0 : part[t - 1];
  for (int i = beg; i < end; ++i) { rowptr[i] = run; run += deg[i]; }
  if (t == 1023) rowptr[n] = part[1023];
}

__global__ __launch_bounds__(256) void k_fill(const int* __restrict__ src,
                                              const int* __restrict__ dst,
                                              const int* __restrict__ rowptr,
                                              int* __restrict__ cursor,
                                              int* __restrict__ col, int e) {
  int i = blockIdx.x * blockDim.x + threadIdx.x;
  if (i < e) {
    int d = dst[i];
    int pos = rowptr[d] + atomicAdd(&cursor[d], 1);
    col[pos] = src[i];
  }
}

// ---------------------------------------------------------------------------
// f32 WMMA GEMM: H[M,N] = X[M,K] * W[K,N], one 16-row tile per block,
// one wave per 16-col tile. M must be a multiple of 16 (50000 = 3125*16).
// ---------------------------------------------------------------------------
template <int K, int N>
__global__ __launch_bounds__((N / 16) * 32) void k_gemm(const float* __restrict__ X,
                                                        const float* __restrict__ W,
                                                        float* __restrict__ H) {
  __shared__ float xs[16][K + 2];  // pad rows: stride K+2 breaks 64-bank conflict
  const int T = (N / 16) * 32;
  int t = threadIdx.x;
  int m0 = blockIdx.x * 16;

  // coalesced stage of the 16xK A-tile into LDS
  for (int f = t; f < 16 * K; f += T) {
    int r = f / K, c = f % K;
    xs[r][c] = X[(size_t)(m0 + r) * K + c];
  }
  __syncthreads();

  int lane = t & 31;
  int w    = t >> 5;
  int half = lane >> 4;   // 0: K pair {0,1}, 1: K pair {2,3}
  int lrow = lane & 15;   // M row (A) / N col (B,C,D)
  int n0   = w * 16;

  v8f c = {0.f, 0.f, 0.f, 0.f, 0.f, 0.f, 0.f, 0.f};
#pragma unroll 4
  for (int k = 0; k < K; k += 4) {
    v2f a, b;
    // A 16x4 f32: lanes 0-15 -> M=lane, K={k,k+1}; lanes 16-31 -> K={k+2,k+3}
    a.x = xs[lrow][k + 2 * half];
    a.y = xs[lrow][k + 2 * half + 1];
    // B 4x16 f32: lanes 0-15 -> N=lane, K={k,k+1}; lanes 16-31 -> K={k+2,k+3}
    b.x = W[(size_t)(k + 2 * half) * N + n0 + lrow];
    b.y = W[(size_t)(k + 2 * half + 1) * N + n0 + lrow];
    c = __builtin_amdgcn_wmma_f32_16x16x4_f32(false, a, false, b, (short)0, c,
                                              false, false);
  }

  // D 16x16 f32: lanes 0-15 -> N=lane, M=v; lanes 16-31 -> M=v+8
#pragma unroll
  for (int v = 0; v < 8; ++v) {
    H[(size_t)(m0 + v + 8 * half) * N + n0 + lrow] = c[v];
  }
}

// ---------------------------------------------------------------------------
// Per-node attention projections: s[n] = h[n].a_src, d[n] = h[n].a_dst
// one wave32 per node
// ---------------------------------------------------------------------------
template <int D>
__global__ __launch_bounds__(256) void k_alpha(const float* __restrict__ H,
                                               const float* __restrict__ a_src,
                                               const float* __restrict__ a_dst,
                                               float* __restrict__ S,
                                               float* __restrict__ Dv, int n) {
  int lane = threadIdx.x & 31;
  int wave = threadIdx.x >> 5;
  int node = blockIdx.x * 8 + wave;
  if (node >= n) return;
  float ss = 0.f, dd = 0.f;
#pragma unroll
  for (int j = 0; j < D / 32; ++j) {
    float v = H[(size_t)node * D + lane + 32 * j];
    ss += v * a_src[lane + 32 * j];
    dd += v * a_dst[lane + 32 * j];
  }
#pragma unroll
  for (int off = 16; off > 0; off >>= 1) {
    ss += __shfl_xor(ss, off, 32);
    dd += __shfl_xor(dd, off, 32);
  }
  if (lane == 0) { S[node] = ss; Dv[node] = dd; }
}

// ---------------------------------------------------------------------------
// Segment softmax + weighted aggregation, one wave32 per destination node.
// Self-loop handled explicitly (not in CSR).
// ---------------------------------------------------------------------------
template <int D>
__global__ __launch_bounds__(256) void k_agg(const float* __restrict__ H,
                                             const float* __restrict__ S,
                                             const float* __restrict__ Dv,
                                             const int* __restrict__ rowptr,
                                             const int* __restrict__ col,
                                             const float* __restrict__ bias,
                                             float* __restrict__ out, int n,
                                             int do_relu) {
  int lane = threadIdx.x & 31;
  int wave = threadIdx.x >> 5;
  int node = blockIdx.x * 8 + wave;
  if (node >= n) return;
  int r0 = rowptr[node], r1 = rowptr[node + 1];
  float dn = Dv[node];

  // pass 1: segment max (incl. self-loop)
  float m = lrelu(S[node] + dn);
  for (int i = r0 + lane; i < r1; i += 32)
    m = fmaxf(m, lrelu(S[col[i]] + dn));
#pragma unroll
  for (int off = 16; off > 0; off >>= 1) m = fmaxf(m, __shfl_xor(m, off, 32));

  // pass 2: exp-sum + weighted feature accumulation in registers
  constexpr int NC = D / 32;
  float acc[NC];
#pragma unroll
  for (int j = 0; j < NC; ++j) acc[j] = 0.f;
  float wsum = 0.f;

  for (int i = r0; i < r1; ++i) {
    int s = col[i];
    float wgt = __expf(lrelu(S[s] + dn) - m);
    wsum += wgt;
#pragma unroll
    for (int j = 0; j < NC; ++j)
      acc[j] += wgt * H[(size_t)s * D + lane + 32 * j];
  }
  {  // self loop
    float wgt = __expf(lrelu(S[node] + dn) - m);
    wsum += wgt;
#pragma unroll
    for (int j = 0; j < NC; ++j)
      acc[j] += wgt * H[(size_t)node * D + lane + 32 * j];
  }

  float inv = 1.f / wsum;
#pragma unroll
  for (int j = 0; j < NC; ++j) {
    float o = acc[j] * inv + bias[lane + 32 * j];
    if (do_relu) o = fmaxf(o, 0.f);
    out[(size_t)node * D + lane + 32 * j] = o;
  }
}

// ---------------------------------------------------------------------------
extern "C" void kernel_launch(void* const* d_in, const int* in_sizes, int n_in,
                              void* d_out, int out_size, void* d_ws, size_t ws_size,
                              hipStream_t stream) {
  (void)in_sizes; (void)n_in; (void)out_size; (void)ws_size;
  const float* x   = (const float*)d_in[0];
  const int*   ei  = (const int*)d_in[1];
  const float* W1  = (const float*)d_in[2];
  const float* as1 = (const float*)d_in[3];
  const float* ad1 = (const float*)d_in[4];
  const float* b1  = (const float*)d_in[5];
  const float* W2  = (const float*)d_in[6];
  const float* as2 = (const float*)d_in[7];
  const float* ad2 = (const float*)d_in[8];
  const float* b2  = (const float*)d_in[9];
  float* out = (float*)d_out;

  const int N = N_NODES, E = N_EDGES;
  const int* srcI = ei;
  const int* dstI = ei + E;

  char* p = (char*)d_ws;
  float* h1  = (float*)p; p += (size_t)N * 128 * 4;
  float* h1r = (float*)p; p += (size_t)N * 128 * 4;
  float* h2  = (float*)p; p += (size_t)N * 64 * 4;
  float* s1  = (float*)p; p += (size_t)N * 4;
  float* d1v = (float*)p; p += (size_t)N * 4;
  float* s2  = (float*)p; p += (size_t)N * 4;
  float* d2v = (float*)p; p += (size_t)N * 4;
  int* deg    = (int*)p; p += (size_t)N * 4;
  int* cursor = (int*)p; p += (size_t)N * 4;
  int* rowptr = (int*)p; p += (size_t)(N + 1) * 4;
  int* col    = (int*)p; p += (size_t)E * 4;

  // CSR of incoming edges (recomputed each call; deterministic work)
  k_zero2<<<(N + 255) / 256, 256, 0, stream>>>(deg, cursor, N);
  k_deg<<<(E + 255) / 256, 256, 0, stream>>>(dstI, deg, E);
  k_scan<<<1, 1024, 0, stream>>>(deg, rowptr, N);
  k_fill<<<(E + 255) / 256, 256, 0, stream>>>(srcI, dstI, rowptr, cursor, col, E);

  // layer 1
  k_gemm<256, 128><<<N / 16, 256, 0, stream>>>(x, W1, h1);
  k_alpha<128><<<(N + 7) / 8, 256, 0, stream>>>(h1, as1, ad1, s1, d1v, N);
  k_agg<128><<<(N + 7) / 8, 256, 0, stream>>>(h1, s1, d1v, rowptr, col, b1, h1r, N, 1);

  // layer 2
  k_gemm<128, 64><<<N / 16, 128, 0, stream>>>(h1r, W2, h2);
  k_alpha<64><<<(N + 7) / 8, 256, 0, stream>>>(h2, as2, ad2, s2, d2v, N);
  k_agg<64><<<(N + 7) / 8, 256, 0, stream>>>(h2, s2, d2v, rowptr, col, b2, out, N, 0);
}